// QuantumGenerator_73796128080505
// MI455X (gfx1250) — compile-verified
//
#include <hip/hip_runtime.h>
#include <hip/hip_bf16.h>
#include <math.h>

typedef __attribute__((ext_vector_type(2))) float v2f;
typedef __attribute__((ext_vector_type(8))) float v8f;

// ---------------- complex helpers (setup kernel only) ----------------
struct Cpx { float re, im; };
__device__ __forceinline__ Cpx cmul(Cpx a, Cpx b) {
    return Cpx{a.re * b.re - a.im * b.im, a.re * b.im + a.im * b.re};
}
__device__ __forceinline__ Cpx cadd(Cpx a, Cpx b) { return Cpx{a.re + b.re, a.im + b.im}; }

// ---------------------------------------------------------------------
// Setup kernel: fold the entire variational part (3 layers of
// RZ*RY*RX per wire + CNOT01 + CNOT10) into one constant 4x4 complex
// unitary U, then emit its 8x8 real representation M padded to 16x8
// (rows 8..15 zero) so the main kernel can feed it as the WMMA A-matrix.
// real mapping:  y[2i]   = sum_j  Re(U_ij) x[2j] - Im(U_ij) x[2j+1]
//                y[2i+1] = sum_j  Im(U_ij) x[2j] + Re(U_ij) x[2j+1]
// ---------------------------------------------------------------------
__global__ void qg_setup_kernel(const float* __restrict__ theta, float* __restrict__ Mpad) {
    if (threadIdx.x != 0 || blockIdx.x != 0) return;

    Cpx U[4][4];
    for (int i = 0; i < 4; ++i)
        for (int j = 0; j < 4; ++j)
            U[i][j] = Cpx{(i == j) ? 1.0f : 0.0f, 0.0f};

    for (int l = 0; l < 3; ++l) {
        Cpx g[2][2][2];
        for (int w = 0; w < 2; ++w) {
            const float t0 = theta[(l * 2 + w) * 3 + 0];
            const float t1 = theta[(l * 2 + w) * 3 + 1];
            const float t2 = theta[(l * 2 + w) * 3 + 2];
            const float cx = cosf(0.5f * t0), sx = sinf(0.5f * t0);
            const float cy = cosf(0.5f * t1), sy = sinf(0.5f * t1);
            const float cz = cosf(0.5f * t2), sz = sinf(0.5f * t2);
            Cpx RX[2][2] = {{{cx, 0.f}, {0.f, -sx}}, {{0.f, -sx}, {cx, 0.f}}};
            Cpx RY[2][2] = {{{cy, 0.f}, {-sy, 0.f}}, {{sy, 0.f}, {cy, 0.f}}};
            Cpx RZ[2][2] = {{{cz, -sz}, {0.f, 0.f}}, {{0.f, 0.f}, {cz, sz}}};
            Cpx T[2][2];
            for (int i = 0; i < 2; ++i)
                for (int j = 0; j < 2; ++j)
                    T[i][j] = cadd(cmul(RY[i][0], RX[0][j]), cmul(RY[i][1], RX[1][j]));
            for (int i = 0; i < 2; ++i)
                for (int j = 0; j < 2; ++j)
                    g[w][i][j] = cadd(cmul(RZ[i][0], T[0][j]), cmul(RZ[i][1], T[1][j]));
        }
        // layer matrix K = kron(g0, g1), state index = q0*2 + q1
        Cpx K[4][4];
        for (int i0 = 0; i0 < 2; ++i0)
            for (int i1 = 0; i1 < 2; ++i1)
                for (int j0 = 0; j0 < 2; ++j0)
                    for (int j1 = 0; j1 < 2; ++j1)
                        K[i0 * 2 + i1][j0 * 2 + j1] = cmul(g[0][i0][j0], g[1][i1][j1]);
        // CNOT01 (ctrl q0, tgt q1): swap rows 2,3.  CNOT10: swap rows 1,3.
        for (int j = 0; j < 4; ++j) { Cpx t = K[2][j]; K[2][j] = K[3][j]; K[3][j] = t; }
        for (int j = 0; j < 4; ++j) { Cpx t = K[1][j]; K[1][j] = K[3][j]; K[3][j] = t; }
        // U = K * U
        Cpx NU[4][4];
        for (int i = 0; i < 4; ++i)
            for (int j = 0; j < 4; ++j) {
                Cpx acc{0.f, 0.f};
                for (int k = 0; k < 4; ++k) acc = cadd(acc, cmul(K[i][k], U[k][j]));
                NU[i][j] = acc;
            }
        for (int i = 0; i < 4; ++i)
            for (int j = 0; j < 4; ++j) U[i][j] = NU[i][j];
    }

    // 16x8 padded real matrix (A-matrix of the WMMAs)
    for (int i = 0; i < 16 * 8; ++i) Mpad[i] = 0.0f;
    for (int i = 0; i < 4; ++i)
        for (int j = 0; j < 4; ++j) {
            Mpad[(2 * i) * 8 + 2 * j]         = U[i][j].re;
            Mpad[(2 * i) * 8 + 2 * j + 1]     = -U[i][j].im;
            Mpad[(2 * i + 1) * 8 + 2 * j]     = U[i][j].im;
            Mpad[(2 * i + 1) * 8 + 2 * j + 1] = U[i][j].re;
        }
}

// ---------------------------------------------------------------------
// Main kernel: one wave handles 32 batch elements.
// Per lane: load z (8B), build embedding state x[0..8) (2 sincos),
// exchange halves with shfl_xor(16), run 4x V_WMMA_F32_16X16X4_F32
// (two column tiles x two K-chunks) against the constant M, then each
// lane<16 owns two full 8-component results and stores two float2s.
// Bandwidth bound: 16 B/element -> ~5.5 us at 23.3 TB/s.
// ---------------------------------------------------------------------
__global__ __launch_bounds__(256) void qg_main_kernel(const float* __restrict__ z,
                                                      const float* __restrict__ Mpad,
                                                      float* __restrict__ out,
                                                      int batch) {
    const int lane = threadIdx.x & 31;
    const int wave = threadIdx.x >> 5;
    const int base = (blockIdx.x * 8 + wave) * 32;
    if (base >= batch) return;  // wave-uniform: EXEC stays all-ones

    // ---- constant A-matrix registers (16x4 f32 layout) ----
    // lanes 0-15: rows 0-15, K = {0,1}; lanes 16-31: rows 0-15, K = {2,3}
    const int r  = lane & 15;
    const int kb = (lane & 16) ? 2 : 0;
    v2f aK0, aK1;
    aK0.x = Mpad[r * 8 + kb];
    aK0.y = Mpad[r * 8 + kb + 1];
    aK1.x = Mpad[r * 8 + kb + 4];
    aK1.y = Mpad[r * 8 + kb + 5];

    // ---- per-lane embedding state for element base+lane ----
    int b = base + lane;
    int bl = (b < batch) ? b : (batch - 1);
    float2 zz = ((const float2*)z)[bl];
    float c0, s0, c1, s1;
    __sincosf(0.5f * zz.x, &s0, &c0);
    __sincosf(0.5f * zz.y, &s1, &c1);
    // v(z) = [c^2 - i*c*s,  s*c + i*s^2]
    const float v0r0 = c0 * c0, v0i0 = -c0 * s0, v0r1 = s0 * c0, v0i1 = s0 * s0;
    const float v1r0 = c1 * c1, v1i0 = -c1 * s1, v1r1 = s1 * c1, v1i1 = s1 * s1;
    float x0 = v0r0 * v1r0 - v0i0 * v1i0, x1 = v0r0 * v1i0 + v0i0 * v1r0;
    float x2 = v0r0 * v1r1 - v0i0 * v1i1, x3 = v0r0 * v1i1 + v0i0 * v1r1;
    float x4 = v0r1 * v1r0 - v0i1 * v1i0, x5 = v0r1 * v1i0 + v0i1 * v1r0;
    float x6 = v0r1 * v1r1 - v0i1 * v1i1, x7 = v0r1 * v1i1 + v0i1 * v1r1;

    // ---- cross-half exchange (lane ^ 16) for B-matrix K placement ----
    const float sx0 = __shfl_xor(x0, 16, 32), sx1 = __shfl_xor(x1, 16, 32);
    const float sx2 = __shfl_xor(x2, 16, 32), sx3 = __shfl_xor(x3, 16, 32);
    const float sx4 = __shfl_xor(x4, 16, 32), sx5 = __shfl_xor(x5, 16, 32);
    const float sx6 = __shfl_xor(x6, 16, 32), sx7 = __shfl_xor(x7, 16, 32);

    const bool lo = (lane & 16) == 0;
    // B 4x16 layout: VGPR0 = K0 (lanes 0-15) / K2 (lanes 16-31); VGPR1 = K1 / K3
    v2f bA_k03, bA_k47, bB_k03, bB_k47;
    bA_k03.x = lo ? x0 : sx2;  bA_k03.y = lo ? x1 : sx3;   // cols = elems base..base+15
    bA_k47.x = lo ? x4 : sx6;  bA_k47.y = lo ? x5 : sx7;
    bB_k03.x = lo ? sx0 : x2;  bB_k03.y = lo ? sx1 : x3;   // cols = elems base+16..base+31
    bB_k47.x = lo ? sx4 : x6;  bB_k47.y = lo ? sx5 : x7;

    // ---- D = M * X via f32 WMMA (exact f32 math) ----
    v8f accA = {0.f, 0.f, 0.f, 0.f, 0.f, 0.f, 0.f, 0.f};
    v8f accB = {0.f, 0.f, 0.f, 0.f, 0.f, 0.f, 0.f, 0.f};
    accA = __builtin_amdgcn_wmma_f32_16x16x4_f32(false, aK0, false, bA_k03, (short)0, accA, false, false);
    accA = __builtin_amdgcn_wmma_f32_16x16x4_f32(false, aK1, false, bA_k47, (short)0, accA, false, false);
    accB = __builtin_amdgcn_wmma_f32_16x16x4_f32(false, aK0, false, bB_k03, (short)0, accB, false, false);
    accB = __builtin_amdgcn_wmma_f32_16x16x4_f32(false, aK1, false, bB_k47, (short)0, accB, false, false);

    // ---- lane n (<16) holds D[0..7][n] = all 8 components of its element ----
    if (lo) {
        {
            const int e = base + lane;
            if (e < batch) {
                float p0 = accA[0] * accA[0] + accA[1] * accA[1];
                float p1 = accA[2] * accA[2] + accA[3] * accA[3];
                float p2 = accA[4] * accA[4] + accA[5] * accA[5];
                float p3 = accA[6] * accA[6] + accA[7] * accA[7];
                float2 o;
                o.x = (p0 + p1) - (p2 + p3);
                o.y = (p0 + p2) - (p1 + p3);
                ((float2*)out)[e] = o;
            }
        }
        {
            const int e = base + 16 + lane;
            if (e < batch) {
                float p0 = accB[0] * accB[0] + accB[1] * accB[1];
                float p1 = accB[2] * accB[2] + accB[3] * accB[3];
                float p2 = accB[4] * accB[4] + accB[5] * accB[5];
                float p3 = accB[6] * accB[6] + accB[7] * accB[7];
                float2 o;
                o.x = (p0 + p1) - (p2 + p3);
                o.y = (p0 + p2) - (p1 + p3);
                ((float2*)out)[e] = o;
            }
        }
    }
}

extern "C" void kernel_launch(void* const* d_in, const int* in_sizes, int n_in,
                              void* d_out, int out_size, void* d_ws, size_t ws_size,
                              hipStream_t stream) {
    const float* z_batch = (const float*)d_in[0];  // [B, 2] f32
    const float* theta   = (const float*)d_in[1];  // [3, 2, 3] f32
    float* Mpad = (float*)d_ws;                    // 16x8 f32 = 512 B scratch
    float* out  = (float*)d_out;                   // [B, 2] f32

    const int batch = in_sizes[0] / 2;

    qg_setup_kernel<<<1, 32, 0, stream>>>(theta, Mpad);

    const int elems_per_block = 256;  // 8 waves * 32 elements
    const int blocks = (batch + elems_per_block - 1) / elems_per_block;
    qg_main_kernel<<<blocks, 256, 0, stream>>>(z_batch, Mpad, out, batch);
}